// RegionConfusionMechanism_30958124270140
// MI455X (gfx1250) — compile-verified
//
#include <hip/hip_runtime.h>
#include <stdint.h>

// RegionConfusionMechanism for MI455X (gfx1250):
// pure data-movement gather of 56x56x(4ch) fp32 tiles, done via the CDNA5
// async global<->LDS DMA path (ASYNCcnt) instead of VGPR round-trips.

constexpr int NREG    = 4;           // N x N region grid
constexpr int C_TOT   = 64;
constexpr int Hh      = 224;
constexpr int Ww      = 224;
constexpr int RH      = 56;          // region height
constexpr int RW      = 56;          // region width
constexpr int CG      = 4;           // channels per workgroup
constexpr int ROW_F4  = RW / 4;      // 14 float4 per region row
constexpr int CH_F4   = RH * ROW_F4; // 784 float4 per channel tile
constexpr int TILE_F4 = CG * CH_F4;  // 3136 float4 per workgroup (50,176 B)
constexpr int THREADS = 256;         // 8 waves (wave32)

__global__ __launch_bounds__(THREADS)
void region_permute_async_kernel(const float* __restrict__ x,
                                 const int*   __restrict__ perm,
                                 float*       __restrict__ out)
{
    __shared__ float4 smem[TILE_F4];   // 50,176 B staging buffer in LDS

    const int c8  = blockIdx.x;        // channel group [0,16)
    const int t   = blockIdx.y;        // target region [0,16)
    const int b   = blockIdx.z;        // batch
    const int tid = threadIdx.x;

    const int src = perm[b * (NREG * NREG) + t];   // block-uniform (scalar load)
    const int ti  = t >> 2, tj = t & 3;

    float* dstBase = out + (((size_t)(b * C_TOT + c8 * CG) * Hh + ti * RH) * Ww + tj * RW);

    if (src < 0) {
        // target slot never written by the scatter -> zeros
        const float4 z = make_float4(0.f, 0.f, 0.f, 0.f);
        for (int i = tid; i < TILE_F4; i += THREADS) {
            int cl  = i / CH_F4;
            int rem = i - cl * CH_F4;
            int r   = rem / ROW_F4;
            int c4  = rem - r * ROW_F4;
            *reinterpret_cast<float4*>(dstBase + ((size_t)cl * Hh + r) * Ww + c4 * 4) = z;
        }
        return;
    }

    const int si = src >> 2, sj = src & 3;
    const float* srcBase = x + (((size_t)(b * C_TOT + c8 * CG) * Hh + si * RH) * Ww + sj * RW);

    // low 32 bits of a generic pointer to __shared__ == LDS byte offset
    const unsigned ldsBase = (unsigned)(uintptr_t)(&smem[0]);

    // Phase 1: pipeline async DMA loads global -> LDS (<=13 outstanding/wave)
    for (int i = tid; i < TILE_F4; i += THREADS) {
        int cl  = i / CH_F4;
        int rem = i - cl * CH_F4;
        int r   = rem / ROW_F4;
        int c4  = rem - r * ROW_F4;
        const float* sp = srcBase + ((size_t)cl * Hh + r) * Ww + c4 * 4;
        unsigned lds = ldsBase + (unsigned)i * 16u;
        asm volatile("global_load_async_to_lds_b128 %0, %1, off"
                     :: "v"(lds), "v"(sp)
                     : "memory");
    }

    // wait for all async loads to land in LDS
    asm volatile("s_wait_asynccnt 0" ::: "memory");

    // Phase 2: pipeline async DMA stores LDS -> global.
    // Same thread wrote each LDS slot it reads -> no barrier needed.
    // s_endpgm's implicit wait-idle drains the stores.
    for (int i = tid; i < TILE_F4; i += THREADS) {
        int cl  = i / CH_F4;
        int rem = i - cl * CH_F4;
        int r   = rem / ROW_F4;
        int c4  = rem - r * ROW_F4;
        float* dp = dstBase + ((size_t)cl * Hh + r) * Ww + c4 * 4;
        unsigned lds = ldsBase + (unsigned)i * 16u;
        asm volatile("global_store_async_from_lds_b128 %0, %1, off"
                     :: "v"(dp), "v"(lds)
                     : "memory");
    }
}

extern "C" void kernel_launch(void* const* d_in, const int* in_sizes, int n_in,
                              void* d_out, int out_size, void* d_ws, size_t ws_size,
                              hipStream_t stream) {
    const float* x    = (const float*)d_in[0];
    const int*   perm = (const int*)d_in[1];
    float*       out  = (float*)d_out;

    const int B = in_sizes[1] / (NREG * NREG);   // 32

    dim3 grid(C_TOT / CG, NREG * NREG, B);       // (16, 16, 32) = 8192 WGs
    region_permute_async_kernel<<<grid, THREADS, 0, stream>>>(x, perm, out);
}